// SimpleMLP_20504173871679
// MI455X (gfx1250) — compile-verified
//
#include <hip/hip_runtime.h>

// CDNA5 / gfx1250, wave32. FFT-butterfly MLP as three radix-16 complex GEMM
// passes per layer via V_WMMA_F32_16X16X4_F32, fully fused in LDS
// (8 rows x 4096 complex interleaved = 256KB/WGP), XOR bank swizzle.

typedef float v2f __attribute__((ext_vector_type(2)));
typedef float v8f __attribute__((ext_vector_type(8)));

#define NFFT     4096
#define LOG2N    12
#define ROWS     8
#define NTHREADS 512
#define NWAVES   (NTHREADS / 32)
#define MATS_PER_LAYER 273      // 1 (pass1) + 16 (pass2) + 256 (pass3)
#define MAT_FLOATS     512      // Ar[256], Ai[256]
#define PI_F 3.14159265358979323846f

// ---------------------------------------------------------------------------
// Prologue: build the 16x16 complex pass matrices from w (twiddle scales).
// Stage S=2^(b+u+1) pairs digit-bit u; twiddle index j = L + (d&(2^u-1))<<b,
// k = j * N/S, tw = exp(-2*pi*i/N * k * w[k]).
// ---------------------------------------------------------------------------
__global__ __launch_bounds__(256) void build_mats(
    const float* __restrict__ w1, const float* __restrict__ w2,
    float* __restrict__ mats)
{
  const int layer = blockIdx.x / MATS_PER_LAYER;
  const int mid   = blockIdx.x % MATS_PER_LAYER;
  const float* w = layer ? w2 : w1;

  int b, L;
  if (mid == 0)      { b = 0; L = 0;        }   // pass 1
  else if (mid < 17) { b = 4; L = mid - 1;  }   // pass 2, L = i
  else               { b = 8; L = mid - 17; }   // pass 3, L = i + 16*m

  __shared__ float mr[2][16][16];
  __shared__ float mi[2][16][16];
  const int t = threadIdx.x;
  const int d = t >> 4;      // output digit (row)
  const int e = t & 15;      // input digit (col)
  mr[0][d][e] = (d == e) ? 1.0f : 0.0f;
  mi[0][d][e] = 0.0f;
  __syncthreads();

  int cur = 0;
  for (int u = 0; u < 4; ++u) {
    const int bit  = 1 << u;
    const int dlow = d & (bit - 1);
    const int j    = L + (dlow << b);
    const int k    = j << (11 - b - u);          // j * (N / 2^(b+u+1))
    const float ang = (-2.0f * PI_F / (float)NFFT) * (float)k * w[k];
    const float tc = cosf(ang);
    const float ts = sinf(ang);

    const int dl = d & ~bit, dh = d | bit;
    const float ar = mr[cur][dl][e], ai = mi[cur][dl][e];
    const float br = mr[cur][dh][e], bi = mi[cur][dh][e];
    const float tr = tc * br - ts * bi;
    const float ti = tc * bi + ts * br;
    float nr, ni;
    if (d & bit) { nr = ar - tr; ni = ai - ti; }
    else         { nr = ar + tr; ni = ai + ti; }
    mr[cur ^ 1][d][e] = nr;
    mi[cur ^ 1][d][e] = ni;
    __syncthreads();
    cur ^= 1;
  }

  float* o = mats + (size_t)(layer * MATS_PER_LAYER + mid) * MAT_FLOATS;
  o[d * 16 + e]       = mr[cur][d][e];
  o[256 + d * 16 + e] = mi[cur][d][e];
}

// ---------------------------------------------------------------------------
// LDS address swizzle. Element (r, n) stored as float2 at LC[cswz(r,n)].
// XOR masks (onto the 5 bank-pair bits, 8B units) chosen so that within each
// WMMA B-load instruction the varying digits map injectively onto banks:
//   pass1 varies (m[3:0], h=i1): naturals m0->bp4, h->bp1; m1->{bp0,bp4},
//         m2->bp2, m3->bp3
//   pass2 varies (G[3:0], h=m1): G->bp[3:0], m1->{bp0,bp4}
//   pass3 varies (r[2:0], G1):   r->bp[4:2], G1->bp1
// ---------------------------------------------------------------------------
__device__ __forceinline__ int cswz(int r, int n)
{
  int q = ((n >> 5) & 1) * 0x11;   // m1 -> bits 0,4
  q ^= (n >> 4) & 0x0C;            // m3,m2 -> bits 3,2
  q ^= (n >> 8) & 0x0F;            // G -> bits 3..0
  q ^= (r & 7) << 2;               // r -> bits 4..2
  return (r << LOG2N) + (n ^ q);
}

// ---------------------------------------------------------------------------
// One complex 16x16x16 product via 16 x V_WMMA_F32_16X16X4_F32.
//   Yre = Ar*Xr + (-Ai)*Xi ,  Yim = Ar*Xi + Ai*Xr
// A layout (ISA 16x4 f32): lane = h*16+d, component v -> A[d][4kc+2h+v].
// B layout (4x16, mirrored): component v -> X[4kc+2h+v][n], n = lane&15.
// nbase/estride locate digit e in the n-index; exor folds the initial
// bit-11 permutation (pass 3 only, exor=8).
// ---------------------------------------------------------------------------
__device__ __forceinline__ void cgemm16(
    const float* __restrict__ mat,
    const float2* lc, int r, int nbase, int estride, int exor,
    v8f& cr, v8f& ci)
{
  const int lane = threadIdx.x & 31;
  const int h = lane >> 4;
  const int d = lane & 15;
#pragma unroll
  for (int z = 0; z < 8; ++z) { cr[z] = 0.0f; ci[z] = 0.0f; }
#pragma unroll
  for (int kc = 0; kc < 4; ++kc) {
    const int k0  = kc * 4 + 2 * h;
    const int ai0 = d * 16 + k0;
    v2f aR = { mat[ai0],       mat[ai0 + 1] };
    v2f aI = { mat[256 + ai0], mat[256 + ai0 + 1] };
    v2f aM = -aI;
    const int e0 = k0 ^ exor;                 // exor=8, k0 even -> e0,e0+1 adjacent
    const float2 b0 = lc[cswz(r, nbase + e0 * estride)];
    const float2 b1 = lc[cswz(r, nbase + (e0 + 1) * estride)];
    v2f bR = { b0.x, b1.x };
    v2f bI = { b0.y, b1.y };
    cr = __builtin_amdgcn_wmma_f32_16x16x4_f32(false, aR, false, bR, (short)0, cr, false, false);
    cr = __builtin_amdgcn_wmma_f32_16x16x4_f32(false, aM, false, bI, (short)0, cr, false, false);
    ci = __builtin_amdgcn_wmma_f32_16x16x4_f32(false, aR, false, bI, (short)0, ci, false, false);
    ci = __builtin_amdgcn_wmma_f32_16x16x4_f32(false, aI, false, bR, (short)0, ci, false, false);
  }
}

// ---------------------------------------------------------------------------
// Fused kernel: load 8 rows, [pass1, pass2, pass3, ReLU] x layer1,
// [pass1, pass2, pass3->out] x layer2, all in LDS.
// ---------------------------------------------------------------------------
__global__ __launch_bounds__(NTHREADS) void fft_mlp(
    const float* __restrict__ x,
    const float* __restrict__ mats,
    float* __restrict__ out)
{
  __shared__ float2 LC[ROWS * NFFT];          // 256 KB, interleaved complex

  const int wave = threadIdx.x >> 5;
  const int lane = threadIdx.x & 31;
  const int h    = lane >> 4;
  const int n16  = lane & 15;
  const size_t row0 = (size_t)blockIdx.x * ROWS;

  // Load input: z[n] = x[2n] + i*x[2n+1] -> swizzled complex store
  for (int idx = threadIdx.x; idx < ROWS * NFFT; idx += NTHREADS) {
    const int r = idx >> LOG2N;
    const int n = idx & (NFFT - 1);
    LC[cswz(r, n)] = ((const float2*)x)[(row0 + r) * NFFT + n];
  }
  __syncthreads();

  v8f cr, ci;
  for (int layer = 0; layer < 2; ++layer) {
    const float* ml = mats + (size_t)layer * MATS_PER_LAYER * MAT_FLOATS;

    // ---- pass 1: digit i (stride 1); columns = (r, G, m); 128 tiles ----
    for (int t = wave; t < ROWS * 16; t += NWAVES) {
      const int c = t * 16 + n16;             // [0, ROWS*256)
      const int r = c >> 8;
      const int G = (c >> 4) & 15;            // constant within tile
      const int m = c & 15;                   // = n16, varies across lanes
      const int nb = (m << 4) + (G << 8);
      cgemm16(ml, LC, r, nb, 1, 0, cr, ci);
#pragma unroll
      for (int rr = 0; rr < 8; ++rr) {
        const int dd = rr + 8 * h;
        LC[cswz(r, nb + dd)] = make_float2(cr[rr], ci[rr]);
      }
    }
    __syncthreads();

    // ---- pass 2: digit m (stride 16); matrices A2[i]; 128 tiles ----
    for (int t = wave; t < 16 * ROWS; t += NWAVES) {
      const int i = t >> 3;                   // 8 tiles per i
      const int c = (t & 7) * 16 + n16;       // [0, 128)
      const int r = c >> 4;                   // constant within tile
      const int G = c & 15;                   // = n16
      const int nb = i + (G << 8);
      cgemm16(ml + (size_t)(1 + i) * MAT_FLOATS, LC, r, nb, 16, 0, cr, ci);
#pragma unroll
      for (int rr = 0; rr < 8; ++rr) {
        const int dd = rr + 8 * h;
        LC[cswz(r, nb + (dd << 4))] = make_float2(cr[rr], ci[rr]);
      }
    }
    __syncthreads();

    // ---- pass 3: digit G (stride 256); matrices A3[i+16m]; perm -> e^8 ----
    for (int t = wave; t < 256; t += NWAVES) {
      const int i = t & 15;
      const int m = t >> 4;
      const int r = n16 & (ROWS - 1);         // cols 8..15 duplicate rows 0..7
      const int nb = i + (m << 4);
      cgemm16(ml + (size_t)(17 + t) * MAT_FLOATS, LC, r, nb, 256, 8, cr, ci);
      if (layer == 0) {
        if (n16 < ROWS) {
#pragma unroll
          for (int rr = 0; rr < 8; ++rr) {
            const int dd = rr + 8 * h;
            LC[cswz(r, nb + (dd << 8))] = make_float2(cr[rr], ci[rr]);
          }
        }
      } else {
        // keep only n_out = i + 16m + 256*G with G < 4  (truncate to 1024)
        if (n16 < ROWS && h == 0) {
          float2* orow = (float2*)(out + (row0 + r) * 2048);
#pragma unroll
          for (int rr = 0; rr < 4; ++rr) {
            const int nout = i + (m << 4) + (rr << 8);
            orow[nout] = make_float2(cr[rr], ci[rr]);
          }
        }
      }
    }
    __syncthreads();

    if (layer == 0) {
      // ReLU elementwise on re and im parts (layout-agnostic)
      float* f = (float*)LC;
      for (int idx = threadIdx.x; idx < 2 * ROWS * NFFT; idx += NTHREADS)
        f[idx] = fmaxf(f[idx], 0.0f);
      __syncthreads();
    }
  }
}

// ---------------------------------------------------------------------------
extern "C" void kernel_launch(void* const* d_in, const int* in_sizes, int n_in,
                              void* d_out, int out_size, void* d_ws, size_t ws_size,
                              hipStream_t stream)
{
  (void)n_in; (void)out_size; (void)ws_size;
  const float* x  = (const float*)d_in[0];
  const float* w1 = (const float*)d_in[1];
  const float* w2 = (const float*)d_in[2];
  float* out  = (float*)d_out;
  float* mats = (float*)d_ws;   // 2 * 273 * 512 floats = ~1.1 MB

  build_mats<<<2 * MATS_PER_LAYER, 256, 0, stream>>>(w1, w2, mats);

  const int nrows = in_sizes[0] / (2 * NFFT);   // BATCH = 4096
  fft_mlp<<<nrows / ROWS, NTHREADS, 0, stream>>>(x, mats, out);
}